// myLSTMCell_386547056754
// MI455X (gfx1250) — compile-verified
//
#include <hip/hip_runtime.h>

// ---------------------------------------------------------------------------
// Peephole LSTM on MI455X / gfx1250 (wave32, fp32 WMMA 16x16x4).
//
//  prep   : transpose + zero-pad all weights to N-major (B frags -> b64),
//           copy h0/c0, reset device barrier.
//  phase A: xg[g][t*B+b][n] = x @ W_xg + b_g  (WMMA GEMM, 4 gates / A reuse)
//  phase B: sequential recurrence, 16 persistent WGs (16-col slice each),
//           device-wide software barrier 2x per timestep.
// ---------------------------------------------------------------------------

#define T_STEPS 512
#define BATCH   64
#define NIN     750
#define KPAD    752          // NIN padded to a multiple of 4
#define H       256
#define NB      16           // recurrence workgroups (H / 16 columns each)
#define MROWS   (T_STEPS * BATCH)   // 32768

typedef __attribute__((ext_vector_type(2))) float v2f;
typedef __attribute__((ext_vector_type(8))) float v8f;

static __device__ __forceinline__ v8f wmma_f32(v2f a, v2f b, v8f c) {
  return __builtin_amdgcn_wmma_f32_16x16x4_f32(false, a, false, b, (short)0, c,
                                               false, false);
}

static __device__ __forceinline__ float sigm(float x) {
  return 1.0f / (1.0f + __expf(-x));
}

// ---------------------------------------------------------------------------
// prep 1: transpose+pad the 4 input weights: Wxt[g][n][k] (k padded to 752)
// ---------------------------------------------------------------------------
__global__ void pack_xw_kernel(const float* __restrict__ Wxi,
                               const float* __restrict__ Wxf,
                               const float* __restrict__ Wxc,
                               const float* __restrict__ Wxo,
                               float* __restrict__ Wxt) {
  const float* W[4] = {Wxi, Wxf, Wxc, Wxo};
  const int total = 4 * H * KPAD;
  for (int idx = blockIdx.x * blockDim.x + threadIdx.x; idx < total;
       idx += gridDim.x * blockDim.x) {
    const int g   = idx / (H * KPAD);
    const int rem = idx - g * (H * KPAD);
    const int n   = rem / KPAD;
    const int k   = rem - n * KPAD;
    Wxt[idx] = (k < NIN) ? W[g][(size_t)k * H + n] : 0.0f;
  }
}

// ---------------------------------------------------------------------------
// prep 2: transpose the 7 recurrent weights: Wht[m][n][k] (256x256 each)
// order: Whi, Whf, Whc, Who, Wci, Wcf, Wco
// ---------------------------------------------------------------------------
__global__ void pack_hw_kernel(const float* __restrict__ Whi,
                               const float* __restrict__ Whf,
                               const float* __restrict__ Whc,
                               const float* __restrict__ Who,
                               const float* __restrict__ Wci,
                               const float* __restrict__ Wcf,
                               const float* __restrict__ Wco,
                               float* __restrict__ Wht) {
  const float* W[7] = {Whi, Whf, Whc, Who, Wci, Wcf, Wco};
  const int total = 7 * H * H;
  for (int idx = blockIdx.x * blockDim.x + threadIdx.x; idx < total;
       idx += gridDim.x * blockDim.x) {
    const int m   = idx / (H * H);
    const int rem = idx - m * (H * H);
    const int n   = rem / H;
    const int k   = rem - n * H;
    Wht[idx] = W[m][(size_t)k * H + n];
  }
}

// ---------------------------------------------------------------------------
// prep 3: copy h0/c0, reset device-wide barrier state.
// ---------------------------------------------------------------------------
__global__ void lstm_init_kernel(const float* __restrict__ h0,
                                 const float* __restrict__ c0,
                                 float* __restrict__ hinit,
                                 float* __restrict__ cbuf,
                                 unsigned* __restrict__ bar) {
  int i = blockIdx.x * blockDim.x + threadIdx.x;
  if (i < BATCH * H) {
    hinit[i] = h0[i];
    cbuf[i]  = c0[i];
  }
  if (i == 0) { bar[0] = 0u; bar[1] = 0u; }
}

// ---------------------------------------------------------------------------
// Phase A: input projections, bias fused. grid=(256,16), 8 waves/block.
// Each wave: one 16x16 M-tile, 4 gate accumulators sharing the A fragment.
// All loads unconditional b64 (weights zero-padded, A address clamped).
// ---------------------------------------------------------------------------
__global__ __launch_bounds__(256) void xproj_kernel(
    const float* __restrict__ x, const float* __restrict__ Wxt,
    const float* __restrict__ bi, const float* __restrict__ bf,
    const float* __restrict__ bc, const float* __restrict__ bo,
    float* __restrict__ xg) {
  const int lane  = threadIdx.x & 31;
  const int wave  = __builtin_amdgcn_readfirstlane(threadIdx.x >> 5);
  const int mtile = blockIdx.x * 8 + wave;      // 0..2047
  const int n0    = blockIdx.y * 16;
  const int row0  = mtile * 16;
  const int l15   = lane & 15;
  const int kb    = (lane >> 4) << 1;           // 0 or 2
  const int mhalf = (lane >> 4) << 3;           // 0 or 8
  const int n     = n0 + l15;

  const float* arow = x + (size_t)(row0 + l15) * NIN;
  const float* brow0 = Wxt + 0 * (size_t)(H * KPAD) + (size_t)n * KPAD;
  const float* brow1 = Wxt + 1 * (size_t)(H * KPAD) + (size_t)n * KPAD;
  const float* brow2 = Wxt + 2 * (size_t)(H * KPAD) + (size_t)n * KPAD;
  const float* brow3 = Wxt + 3 * (size_t)(H * KPAD) + (size_t)n * KPAD;

  v8f acc0 = {}, acc1 = {}, acc2 = {}, acc3 = {};

#pragma unroll 2
  for (int k0 = 0; k0 < KPAD; k0 += 4) {
    const int k  = k0 + kb;
    const int kc = (k < NIN) ? k : (NIN - 2);   // clamp; padded B zeroes tail
    v2f a, b0, b1, b2, b3;
    a.x  = arow[kc];      a.y  = arow[kc + 1];
    b0.x = brow0[k];      b0.y = brow0[k + 1];
    b1.x = brow1[k];      b1.y = brow1[k + 1];
    b2.x = brow2[k];      b2.y = brow2[k + 1];
    b3.x = brow3[k];      b3.y = brow3[k + 1];
    acc0 = wmma_f32(a, b0, acc0);
    acc1 = wmma_f32(a, b1, acc1);
    acc2 = wmma_f32(a, b2, acc2);
    acc3 = wmma_f32(a, b3, acc3);
  }

  const float bias0 = bi[n], bias1 = bf[n], bias2 = bc[n], bias3 = bo[n];
  float* og0 = xg + 0 * (size_t)MROWS * H;
  float* og1 = xg + 1 * (size_t)MROWS * H;
  float* og2 = xg + 2 * (size_t)MROWS * H;
  float* og3 = xg + 3 * (size_t)MROWS * H;
#pragma unroll
  for (int v = 0; v < 8; ++v) {
    const size_t m = (size_t)(row0 + v + mhalf);
    og0[m * H + n] = acc0[v] + bias0;
    og1[m * H + n] = acc1[v] + bias1;
    og2[m * H + n] = acc2[v] + bias2;
    og3[m * H + n] = acc3[v] + bias3;
  }
}

// ---------------------------------------------------------------------------
// Device-wide software barrier (16 co-resident blocks).
// ---------------------------------------------------------------------------
static __device__ __forceinline__ void grid_barrier(unsigned* cnt, unsigned* ep,
                                                    unsigned my_epoch) {
  __syncthreads();
  if (threadIdx.x == 0) {
    __threadfence();
    unsigned prev = atomicAdd(cnt, 1u);
    if (prev == NB - 1) {
      __hip_atomic_store(cnt, 0u, __ATOMIC_RELAXED, __HIP_MEMORY_SCOPE_AGENT);
      __hip_atomic_store(ep, my_epoch + 1u, __ATOMIC_RELEASE,
                         __HIP_MEMORY_SCOPE_AGENT);
    } else {
      while (__hip_atomic_load(ep, __ATOMIC_ACQUIRE,
                               __HIP_MEMORY_SCOPE_AGENT) <= my_epoch)
        __builtin_amdgcn_s_sleep(1);
    }
    __threadfence();
  }
  __syncthreads();
}

// ---------------------------------------------------------------------------
// Phase B: recurrence. 16 blocks x 256 thr. Weights are pre-transposed
// (Wht, N-major -> B fragments are single b64 loads). h history lives in
// d_out (h_{t-1} = out[t-1]); c is ping-ponged in ws.
// ---------------------------------------------------------------------------
__global__ __launch_bounds__(256) void lstm_rec_kernel(
    const float* __restrict__ Wht,     // [7][H][H] transposed (i,f,c,o,ci,cf,co)
    const float* __restrict__ xg,      // [4][T*B][H]
    const float* __restrict__ hinit,   // [B][H]
    float* __restrict__ cbuf,          // [2][B][H]
    float* __restrict__ out,           // [T][B][H]
    unsigned* __restrict__ bar) {
  __shared__ float st_i[BATCH * 16];
  __shared__ float st_f[BATCH * 16];
  __shared__ float st_g[BATCH * 16];
  __shared__ float st_o0[BATCH * 16];
  __shared__ float st_o1[BATCH * 16];
  __shared__ float st_c[BATCH * 16];

  const int lane  = threadIdx.x & 31;
  const int wave  = __builtin_amdgcn_readfirstlane(threadIdx.x >> 5);
  const int n0    = blockIdx.x * 16;
  const int l15   = lane & 15;
  const int kb    = (lane >> 4) << 1;
  const int mhalf = (lane >> 4) << 3;

  // transposed weight rows for this wave's output column (n0 + l15)
  const size_t wrow = (size_t)(n0 + l15) * H;
  const float* Wt_i  = Wht + 0 * (size_t)(H * H) + wrow;
  const float* Wt_f  = Wht + 1 * (size_t)(H * H) + wrow;
  const float* Wt_c  = Wht + 2 * (size_t)(H * H) + wrow;
  const float* Wt_o  = Wht + 3 * (size_t)(H * H) + wrow;
  const float* Wt_ci = Wht + 4 * (size_t)(H * H) + wrow;
  const float* Wt_cf = Wht + 5 * (size_t)(H * H) + wrow;
  const float* Wt_co = Wht + 6 * (size_t)(H * H) + wrow;
  const float* WtH[3] = {Wt_i, Wt_f, Wt_c};
  const float* WtC[2] = {Wt_ci, Wt_cf};

  unsigned* bar_cnt = bar;
  unsigned* bar_ep  = bar + 1;
  unsigned epoch = 0;

  for (int t = 0; t < T_STEPS; ++t) {
    const float* hprev = (t == 0) ? hinit : (out + (size_t)(t - 1) * BATCH * H);
    const float* cin   = cbuf + (size_t)(t & 1) * (BATCH * H);
    float*       cout  = cbuf + (size_t)((t + 1) & 1) * (BATCH * H);

    // ---- phase 1: gates i, f, c~ (12 tile tasks over 8 waves) -------------
    for (int task = wave; task < 12; task += 8) {
      const int g    = task >> 2;               // 0=i, 1=f, 2=c~  (scalar)
      const int row0 = (task & 3) * 16;
      const float* xgate =
          xg + (size_t)g * ((size_t)MROWS * H) + (size_t)t * BATCH * H;
      v8f acc;
#pragma unroll
      for (int v = 0; v < 8; ++v)
        acc[v] = xgate[(size_t)(row0 + v + mhalf) * H + n0 + l15];

      const float* ha = hprev + (size_t)(row0 + l15) * H;
      const float* ca = cin + (size_t)(row0 + l15) * H;
      const float* bw = WtH[g];
      const float* pw = (g < 2) ? WtC[g] : Wt_i; // unused when g==2
      if (g < 2) {
#pragma unroll 4
        for (int k0 = 0; k0 < H; k0 += 4) {
          const int k = k0 + kb;
          v2f ah, bh, ac, bc;
          ah.x = ha[k]; ah.y = ha[k + 1];
          bh.x = bw[k]; bh.y = bw[k + 1];
          ac.x = ca[k]; ac.y = ca[k + 1];
          bc.x = pw[k]; bc.y = pw[k + 1];
          acc = wmma_f32(ah, bh, acc);
          acc = wmma_f32(ac, bc, acc);
        }
      } else {
#pragma unroll 4
        for (int k0 = 0; k0 < H; k0 += 4) {
          const int k = k0 + kb;
          v2f ah, bh;
          ah.x = ha[k]; ah.y = ha[k + 1];
          bh.x = bw[k]; bh.y = bw[k + 1];
          acc = wmma_f32(ah, bh, acc);
        }
      }
      float* st = (g == 0) ? st_i : (g == 1) ? st_f : st_g;
#pragma unroll
      for (int v = 0; v < 8; ++v)
        st[(row0 + v + mhalf) * 16 + l15] = acc[v];
    }
    __syncthreads();

    // ---- elementwise c_t for this slice -----------------------------------
    for (int idx = threadIdx.x; idx < BATCH * 16; idx += 256) {
      const int b = idx >> 4, n = idx & 15;
      const float cnew = sigm(st_f[idx]) * cin[(size_t)b * H + n0 + n] +
                         sigm(st_i[idx]) * tanhf(st_g[idx]);
      st_c[idx] = cnew;
      cout[(size_t)b * H + n0 + n] = cnew;
    }
    grid_barrier(bar_cnt, bar_ep, epoch++);     // full c_t now visible

    // ---- phase 2: o gate (waves 0-3: h@W_ho, waves 4-7: c_t@W_co) ---------
    {
      const int row0 = (wave & 3) * 16;
      const int part = wave >> 2;               // scalar
      v8f acc = {};
      if (part == 0) {
        const float* xgate =
            xg + (size_t)3 * ((size_t)MROWS * H) + (size_t)t * BATCH * H;
#pragma unroll
        for (int v = 0; v < 8; ++v)
          acc[v] = xgate[(size_t)(row0 + v + mhalf) * H + n0 + l15];
      }
      const float* Arow =
          ((part == 0) ? hprev : cout) + (size_t)(row0 + l15) * H;
      const float* bw = (part == 0) ? Wt_o : Wt_co;
#pragma unroll 4
      for (int k0 = 0; k0 < H; k0 += 4) {
        const int k = k0 + kb;
        v2f a, b;
        a.x = Arow[k]; a.y = Arow[k + 1];
        b.x = bw[k];   b.y = bw[k + 1];
        acc = wmma_f32(a, b, acc);
      }
      float* st = part ? st_o1 : st_o0;
#pragma unroll
      for (int v = 0; v < 8; ++v)
        st[(row0 + v + mhalf) * 16 + l15] = acc[v];
    }
    __syncthreads();

    // ---- elementwise h_t, write straight to output ------------------------
    float* outT = out + (size_t)t * BATCH * H;
    for (int idx = threadIdx.x; idx < BATCH * 16; idx += 256) {
      const int b = idx >> 4, n = idx & 15;
      const float o = sigm(st_o0[idx] + st_o1[idx]);
      outT[(size_t)b * H + n0 + n] = o * tanhf(st_c[idx]);
    }
    grid_barrier(bar_cnt, bar_ep, epoch++);     // out[t], cbuf ready for t+1
  }
}

// ---------------------------------------------------------------------------
// Launch. Workspace layout (bytes):
//   0x0000000  xg   : 4 x 32768 x 256 f32          (128 MiB)
//   0x8000000  Wxt  : 4 x 256 x 752 f32            (~3 MiB)
//   0x8400000  Wht  : 7 x 256 x 256 f32            (~1.75 MiB)
//   0x8600000  cbuf : 2 x 64 x 256 f32
//   0x8620000  hinit: 64 x 256 f32
//   0x8630000  bar  : 2 x u32
// ---------------------------------------------------------------------------
extern "C" void kernel_launch(void* const* d_in, const int* in_sizes, int n_in,
                              void* d_out, int out_size, void* d_ws,
                              size_t ws_size, hipStream_t stream) {
  const float* x   = (const float*)d_in[0];
  const float* h0  = (const float*)d_in[1];
  const float* c0  = (const float*)d_in[2];
  const float* Wxi = (const float*)d_in[3];
  const float* Wxf = (const float*)d_in[4];
  const float* Wxc = (const float*)d_in[5];
  const float* Wxo = (const float*)d_in[6];
  const float* Whi = (const float*)d_in[7];
  const float* Whf = (const float*)d_in[8];
  const float* Whc = (const float*)d_in[9];
  const float* Who = (const float*)d_in[10];
  const float* Wci = (const float*)d_in[11];
  const float* Wcf = (const float*)d_in[12];
  const float* Wco = (const float*)d_in[13];
  const float* bi  = (const float*)d_in[14];
  const float* bf  = (const float*)d_in[15];
  const float* bc  = (const float*)d_in[16];
  const float* bo  = (const float*)d_in[17];

  char* ws = (char*)d_ws;
  float*    xg    = (float*)(ws);
  float*    Wxt   = (float*)(ws + (size_t)0x8000000);
  float*    Wht   = (float*)(ws + (size_t)0x8400000);
  float*    cbuf  = (float*)(ws + (size_t)0x8600000);
  float*    hinit = (float*)(ws + (size_t)0x8620000);
  unsigned* bar   = (unsigned*)(ws + (size_t)0x8630000);
  float*    out   = (float*)d_out;

  pack_xw_kernel<<<512, 256, 0, stream>>>(Wxi, Wxf, Wxc, Wxo, Wxt);
  pack_hw_kernel<<<512, 256, 0, stream>>>(Whi, Whf, Whc, Who, Wci, Wcf, Wco,
                                          Wht);
  lstm_init_kernel<<<(BATCH * H + 255) / 256, 256, 0, stream>>>(h0, c0, hinit,
                                                                cbuf, bar);

  dim3 gA(MROWS / (16 * 8), H / 16);            // (256, 16)
  xproj_kernel<<<gA, 256, 0, stream>>>(x, Wxt, bi, bf, bc, bo, xg);

  lstm_rec_kernel<<<NB, 256, 0, stream>>>(Wht, xg, hinit, cbuf, out, bar);
}